// GaussianLinearInvariantSSM_4183298146803
// MI455X (gfx1250) — compile-verified
//
#include <hip/hip_runtime.h>
#include <math.h>

typedef float v2f __attribute__((ext_vector_type(2)));
typedef float v4f __attribute__((ext_vector_type(4)));
typedef float v8f __attribute__((ext_vector_type(8)));

#define D_OBS 8
#define D_STATE 16

static __device__ inline v8f wmma4(v2f a, v2f b, v8f c) {
  // V_WMMA_F32_16X16X4_F32 : D = A(16x4) * B(4x16) + C(16x16)
  return __builtin_amdgcn_wmma_f32_16x16x4_f32(
      /*neg_a=*/false, a, /*neg_b=*/false, b,
      /*c_mod=*/(short)0, c, /*reuse_a=*/false, /*reuse_b=*/false);
}

// ---------------------------------------------------------------------------
// Phase 1: data-independent covariance recursion (runs once, 1 block).
// Produces per-t: SBR_t [16x8], iv_inv_t [8x8], and the scalar constant
// sum_t 0.5*(8*log2pi + logdet(innov_var_t)).
// ---------------------------------------------------------------------------
__global__ __launch_bounds__(256)
void kf_cov_kernel(const float* __restrict__ A, const float* __restrict__ B,
                   const float* __restrict__ U, const float* __restrict__ V,
                   const float* __restrict__ W, const float* __restrict__ A0,
                   float* __restrict__ ws_sbr, float* __restrict__ ws_iv,
                   float* __restrict__ ws_c, int T) {
  __shared__ float sA[256], sQ[256], sS[256], sP[256], sAP[256], sWU[256];
  __shared__ float sB[128], sBS[128], sSBt[128], sSBR[128];
  __shared__ float sV[64], sCH[64], sINV[64];

  const int tid = threadIdx.x;
  const int i16 = tid >> 4, j16 = tid & 15;

  sA[tid] = A[tid];
  sS[tid] = A0[tid];
  if (tid < 128) sB[tid] = B[tid];
  if (tid < 64)  sV[tid] = V[tid];
  { // WU = W @ U   (W: 16x16 since D_NOISE == D_STATE)
    float acc = 0.f;
    for (int k = 0; k < 16; ++k) acc += W[i16 * 16 + k] * U[k * 16 + j16];
    sWU[tid] = acc;
  }
  __syncthreads();
  { // Q = WU @ W^T
    float acc = 0.f;
    for (int k = 0; k < 16; ++k) acc += sWU[i16 * 16 + k] * W[j16 * 16 + k];
    sQ[tid] = acc;
  }
  __syncthreads();

  float c_acc = 0.f;
  const float LOG2PI = 1.8378770664093453f;

  for (int t = 0; t < T; ++t) {
    // BS = B @ S   [8,16]
    if (tid < 128) {
      int i = tid >> 4, j = tid & 15;
      float acc = 0.f;
      for (int k = 0; k < 16; ++k) acc += sB[i * 16 + k] * sS[k * 16 + j];
      sBS[tid] = acc;
    }
    __syncthreads();
    // innov_var = BS @ B^T + V  [8,8]  (stored into sCH for in-place Cholesky)
    if (tid < 64) {
      int i = tid >> 3, j = tid & 7;
      float acc = sV[tid];
      for (int k = 0; k < 16; ++k) acc += sBS[i * 16 + k] * sB[j * 16 + k];
      sCH[tid] = acc;
    }
    __syncthreads();
    // Cholesky (8x8, SPD) by thread 0; accumulates 0.5*logdet = sum log L_ii
    if (tid == 0) {
      float ld = 0.f;
      for (int c = 0; c < 8; ++c) {
        float d = sCH[c * 8 + c];
        for (int k = 0; k < c; ++k) d -= sCH[c * 8 + k] * sCH[c * 8 + k];
        d = sqrtf(d);
        sCH[c * 8 + c] = d;
        ld += logf(d);
        float inv = 1.0f / d;
        for (int r = c + 1; r < 8; ++r) {
          float v = sCH[r * 8 + c];
          for (int k = 0; k < c; ++k) v -= sCH[r * 8 + k] * sCH[c * 8 + k];
          sCH[r * 8 + c] = v * inv;
        }
      }
      c_acc += 0.5f * (8.0f * LOG2PI) + ld;
    }
    __syncthreads();
    // iv_inv: solve L L^T Z = I, one column per thread
    if (tid < 8) {
      const int j = tid;
      float y[8], z[8];
      for (int i = 0; i < 8; ++i) {
        float v = (i == j) ? 1.0f : 0.0f;
        for (int k = 0; k < i; ++k) v -= sCH[i * 8 + k] * y[k];
        y[i] = v / sCH[i * 8 + i];
      }
      for (int i = 7; i >= 0; --i) {
        float v = y[i];
        for (int k = i + 1; k < 8; ++k) v -= sCH[k * 8 + i] * z[k];
        z[i] = v / sCH[i * 8 + i];
      }
      for (int i = 0; i < 8; ++i) sINV[i * 8 + j] = z[i];
    }
    __syncthreads();
    // SBt = S @ B^T  [16,8]
    if (tid < 128) {
      int i = tid >> 3, j = tid & 7;
      float acc = 0.f;
      for (int k = 0; k < 16; ++k) acc += sS[i * 16 + k] * sB[j * 16 + k];
      sSBt[tid] = acc;
    }
    __syncthreads();
    // SBR = SBt @ iv_inv  [16,8]
    if (tid < 128) {
      int i = tid >> 3, j = tid & 7;
      float acc = 0.f;
      for (int k = 0; k < 8; ++k) acc += sSBt[i * 8 + k] * sINV[k * 8 + j];
      sSBR[tid] = acc;
    }
    __syncthreads();
    if (tid < 128) ws_sbr[(size_t)t * 128 + tid] = sSBR[tid];
    if (tid < 64)  ws_iv[(size_t)t * 64 + tid] = sINV[tid];
    // P = S - SBR @ BS  [16,16]
    {
      float acc = sS[tid];
      for (int k = 0; k < 8; ++k) acc -= sSBR[i16 * 8 + k] * sBS[k * 16 + j16];
      sP[tid] = acc;
    }
    __syncthreads();
    // AP = A @ P
    {
      float acc = 0.f;
      for (int k = 0; k < 16; ++k) acc += sA[i16 * 16 + k] * sP[k * 16 + j16];
      sAP[tid] = acc;
    }
    __syncthreads();
    // S = AP @ A^T + Q
    {
      float acc = sQ[tid];
      for (int k = 0; k < 16; ++k) acc += sAP[i16 * 16 + k] * sA[j16 * 16 + k];
      __syncthreads();
      sS[tid] = acc;
    }
    __syncthreads();
  }
  if (tid == 0) ws_c[0] = c_acc;
}

// ---------------------------------------------------------------------------
// Phase 2: batched mean recursion + NLL, WMMA f32 16x16x4.
// One wave per 16-batch tile. Orientation: feature dims on M/K, batch on N
// (lane = batch column). State mu [16 state, 16 batch] round-trips through
// per-wave LDS stored column-major so B-operand K-chunks are aligned v2f loads.
// ---------------------------------------------------------------------------
__global__ __launch_bounds__(32)
void kf_mean_kernel(const float* __restrict__ x, const float* __restrict__ avec,
                    const float* __restrict__ bvec, const float* __restrict__ A,
                    const float* __restrict__ B, const float* __restrict__ a0,
                    const float* __restrict__ ws_sbr,
                    const float* __restrict__ ws_iv,
                    const float* __restrict__ ws_c,
                    float* __restrict__ out, int T) {
  __shared__ float lds_mu[256];   // [col=batch][row=state], col-major
  __shared__ float lds_tmp[256];  // scratch for innov / post
  const int lane = threadIdx.x;
  const int n = lane & 15;        // batch column within tile
  const int h = lane >> 4;        // half (K/M sub-slice select)
  const int m = n;                // A-operand row index
  const int n0 = blockIdx.x * 16;

  // Constant A-operands: negated padded B (rows >= 8 zero) and A, 4 K-chunks.
  v2f nb[4], Ac[4];
#pragma unroll
  for (int kc = 0; kc < 4; ++kc) {
#pragma unroll
    for (int v = 0; v < 2; ++v) {
      const int k = 4 * kc + 2 * h + v;
      nb[kc][v] = (m < D_OBS) ? -B[m * D_STATE + k] : 0.0f;
      Ac[kc][v] = A[m * D_STATE + k];
    }
  }
  // C-operand broadcast of 'a' (a[row]) and the obs bias.
  v8f abc;
  float bv[D_OBS];
#pragma unroll
  for (int r = 0; r < 8; ++r) abc[r] = avec[8 * h + r];
#pragma unroll
  for (int r = 0; r < D_OBS; ++r) bv[r] = bvec[r];

  // mu0 = a0 broadcast over batch columns.
  v8f mu;
#pragma unroll
  for (int r = 0; r < 8; ++r) mu[r] = a0[8 * h + r];
#pragma unroll
  for (int r = 0; r < 8; ++r) lds_mu[n * 16 + 8 * h + r] = mu[r];

  float acc = 0.f;

  for (int t = 0; t < T; ++t) {
    const float* sbrT = ws_sbr + (size_t)t * 128;
    const float* ivT  = ws_iv  + (size_t)t * 64;

    // mu B-operand chunks (col-major LDS -> aligned 8B loads)
    v2f mu0 = *(const v2f*)&lds_mu[n * 16 + 0  + 2 * h];
    v2f mu1 = *(const v2f*)&lds_mu[n * 16 + 4  + 2 * h];
    v2f mu2 = *(const v2f*)&lds_mu[n * 16 + 8  + 2 * h];
    v2f mu3 = *(const v2f*)&lds_mu[n * 16 + 12 + 2 * h];

    // C = x_t - b (rows 0..7 on h==0 lanes), 0 elsewhere
    v8f xc;
    if (h == 0) {
      const float* xb = x + ((size_t)(n0 + n) * (size_t)T + (size_t)t) * D_OBS;
      v4f x0 = *(const v4f*)(xb);
      v4f x1 = *(const v4f*)(xb + 4);
#pragma unroll
      for (int r = 0; r < 4; ++r) { xc[r] = x0[r] - bv[r]; xc[r + 4] = x1[r] - bv[r + 4]; }
    } else {
#pragma unroll
      for (int r = 0; r < 8; ++r) xc[r] = 0.f;
    }

    // innov = (-Bp) @ mu + (x - b)   : 4 chained WMMAs over K=16
    v8f inn = wmma4(nb[0], mu0, xc);
    inn = wmma4(nb[1], mu1, inn);
    inn = wmma4(nb[2], mu2, inn);
    inn = wmma4(nb[3], mu3, inn);

    // stash innov column-major (rows 8..15 are zero)
#pragma unroll
    for (int r = 0; r < 8; ++r) lds_tmp[n * 16 + 8 * h + r] = inn[r];

    // quadratic form: h==0 lanes hold the full nonzero innovation column
    if (h == 0) {
      float q = 0.f;
#pragma unroll
      for (int i = 0; i < 8; ++i) {
        float w = 0.f;
#pragma unroll
        for (int j = 0; j < 8; ++j) w += ivT[i * 8 + j] * inn[j];
        q += inn[i] * w;
      }
      acc += q;
    }

    // post = SBR_t @ innov_pad + mu  : padded K, only chunks 0..1 nonzero
    v2f sa0 = *(const v2f*)(sbrT + m * 8 + 0 + 2 * h);
    v2f sa1 = *(const v2f*)(sbrT + m * 8 + 4 + 2 * h);
    v2f ic0 = *(const v2f*)&lds_tmp[n * 16 + 0 + 2 * h];
    v2f ic1 = *(const v2f*)&lds_tmp[n * 16 + 4 + 2 * h];
    v8f post = wmma4(sa0, ic0, mu);
    post = wmma4(sa1, ic1, post);

    // mu' = A @ post + a_broadcast
#pragma unroll
    for (int r = 0; r < 8; ++r) lds_tmp[n * 16 + 8 * h + r] = post[r];
    v2f pc0 = *(const v2f*)&lds_tmp[n * 16 + 0  + 2 * h];
    v2f pc1 = *(const v2f*)&lds_tmp[n * 16 + 4  + 2 * h];
    v2f pc2 = *(const v2f*)&lds_tmp[n * 16 + 8  + 2 * h];
    v2f pc3 = *(const v2f*)&lds_tmp[n * 16 + 12 + 2 * h];
    mu = wmma4(Ac[0], pc0, abc);
    mu = wmma4(Ac[1], pc1, mu);
    mu = wmma4(Ac[2], pc2, mu);
    mu = wmma4(Ac[3], pc3, mu);
#pragma unroll
    for (int r = 0; r < 8; ++r) lds_mu[n * 16 + 8 * h + r] = mu[r];
  }

  if (h == 0) out[n0 + n] = 0.5f * acc + ws_c[0];
}

extern "C" void kernel_launch(void* const* d_in, const int* in_sizes, int n_in,
                              void* d_out, int out_size, void* d_ws, size_t ws_size,
                              hipStream_t stream) {
  const float* x  = (const float*)d_in[0];
  const float* a  = (const float*)d_in[1];
  const float* b  = (const float*)d_in[2];
  const float* A  = (const float*)d_in[3];
  const float* B  = (const float*)d_in[4];
  const float* U  = (const float*)d_in[5];
  const float* V  = (const float*)d_in[6];
  const float* W  = (const float*)d_in[7];
  const float* a0 = (const float*)d_in[8];
  const float* A0 = (const float*)d_in[9];
  float* out = (float*)d_out;

  const int BATCH = out_size;                       // 256
  const int T = in_sizes[0] / (BATCH * D_OBS);      // 1024

  float* ws = (float*)d_ws;
  float* ws_sbr = ws;                               // T * 128
  float* ws_iv  = ws + (size_t)T * 128;             // T * 64
  float* ws_c   = ws + (size_t)T * 192;             // 1

  kf_cov_kernel<<<1, 256, 0, stream>>>(A, B, U, V, W, A0, ws_sbr, ws_iv, ws_c, T);
  kf_mean_kernel<<<BATCH / 16, 32, 0, stream>>>(x, a, b, A, B, a0,
                                                ws_sbr, ws_iv, ws_c, out, T);
}